// BMFT_48017734369477
// MI455X (gfx1250) — compile-verified
//
#include <hip/hip_runtime.h>
#include <hip/hip_bf16.h>
#include <math.h>

// MoFE for MI455X (gfx1250).
// Roofline: ~0.9 GB min HBM traffic -> ~39us at 23.3 TB/s. Depthwise 3x3 conv
// cannot be a matmul (per-channel filters), so WMMA carries only the gate
// GEMMs (v_wmma_f32_16x16x4_f32). The conv kernel stages x tiles into LDS with
// CDNA5 async global->LDS loads, register-blocks 2x4 outputs per thread to cut
// LDS traffic 3x vs naive, and streams out with nontemporal b128 stores.

#define B_  32
#define C_  64
#define H_  192
#define W_  192
#define E_  6
#define KTOP 3

// conv tile: 32 rows x 64 cols of output per block, 256 threads (8 waves)
#define TH  32
#define TW  64
#define XS_H (TH + 4)     // 36 rows of staged x
#define XS_NW (TW + 4)    // 68 valid cols
#define XS_W 72           // padded row stride (words)
#define HS_H (TH + 2)     // 34 rows of intermediate h
#define HS_NW (TW + 2)    // 66 valid cols
#define HS_W 67           // odd stride => conflict-free half-wave row split

typedef __attribute__((ext_vector_type(2))) float v2f;
typedef __attribute__((ext_vector_type(4))) float v4f;
typedef __attribute__((ext_vector_type(8))) float v8f;

// ---------------------------------------------------------------------------
// Kernel 1: pooled[b,c] = max_{hw} x + mean_{hw} x          (reads 302 MB)
// ---------------------------------------------------------------------------
__global__ __launch_bounds__(256) void mofe_pool_kernel(
    const float* __restrict__ x, float* __restrict__ pooled) {
  const int plane = blockIdx.x;                       // b*C + c
  const float4* p4 = (const float4*)(x + (size_t)plane * (H_ * W_));
  const int tid = threadIdx.x;
  const int n4 = (H_ * W_) / 4;                       // 9216

  float s = 0.f, m = -INFINITY;
  for (int i = tid; i < n4; i += 256) {
    float4 v = p4[i];
    s += (v.x + v.y) + (v.z + v.w);
    m = fmaxf(m, fmaxf(fmaxf(v.x, v.y), fmaxf(v.z, v.w)));
  }
  __shared__ float ss[256];
  __shared__ float sm[256];
  ss[tid] = s; sm[tid] = m;
  __syncthreads();
  for (int st = 128; st > 0; st >>= 1) {
    if (tid < st) {
      ss[tid] += ss[tid + st];
      sm[tid] = fmaxf(sm[tid], sm[tid + st]);
    }
    __syncthreads();
  }
  if (tid == 0) pooled[plane] = sm[0] + ss[0] * (1.0f / (H_ * W_));
}

// ---------------------------------------------------------------------------
// Kernel 2: gate. One wave (32 lanes, EXEC all ones at every WMMA).
// Two GEMMs [32,64]x[64,6->pad16] on v_wmma_f32_16x16x4_f32:
//   A 16x4 f32: lane&15 = M; VGPR0/1 = {K0,K1} (lanes 0-15), {K2,K3} (16-31).
//   B 4x16:     lane&15 = N; VGPR0/1 = {K0,K1} / {K2,K3} likewise.
//   C/D 16x16:  VGPR r, lanes 0-15 -> M=r, lanes 16-31 -> M=r+8.
// ---------------------------------------------------------------------------
__global__ __launch_bounds__(32) void mofe_gate_kernel(
    const float* __restrict__ pooled,
    const float* __restrict__ fc0_w, const float* __restrict__ fc0_b,
    const float* __restrict__ fc1_w, const float* __restrict__ fc1_b,
    float* __restrict__ cof) {
  const int lane = threadIdx.x;
  const int half = lane >> 4;      // 0: K0/K1 ; 1: K2/K3
  const int l15  = lane & 15;

  __shared__ float gbuf[B_][E_];
  __shared__ float nbuf[B_][E_];

  for (int mt = 0; mt < 2; ++mt) {          // two 16-row batch tiles
    v8f accg = {};
    v8f accn = {};
    const int row = mt * 16 + l15;          // batch index for A
    #pragma unroll
    for (int kk = 0; kk < 16; ++kk) {       // K = 64 in steps of 4
      const int kb = kk * 4 + half * 2;
      v2f a, bg, bn;
      a.x = pooled[row * C_ + kb];
      a.y = pooled[row * C_ + kb + 1];
      if (l15 < E_) {
        bg.x = fc1_w[l15 * C_ + kb];  bg.y = fc1_w[l15 * C_ + kb + 1];
        bn.x = fc0_w[l15 * C_ + kb];  bn.y = fc0_w[l15 * C_ + kb + 1];
      } else {
        bg.x = 0.f; bg.y = 0.f; bn.x = 0.f; bn.y = 0.f;
      }
      accg = __builtin_amdgcn_wmma_f32_16x16x4_f32(
          false, a, false, bg, (short)0, accg, false, false);
      accn = __builtin_amdgcn_wmma_f32_16x16x4_f32(
          false, a, false, bn, (short)0, accn, false, false);
    }
    if (l15 < E_) {
      #pragma unroll
      for (int r = 0; r < 8; ++r) {
        const int M = mt * 16 + half * 8 + r;
        gbuf[M][l15] = accg[r];
        nbuf[M][l15] = accn[r];
      }
    }
  }
  __syncthreads();

  // Scalar tail: leaky_relu, softplus, ddof=1 normalize, top-3, masked softmax.
  const int b = lane;
  float g[E_], sp[E_];
  float nsum = 0.f;
  #pragma unroll
  for (int e = 0; e < E_; ++e) {
    float gv = gbuf[b][e] + fc1_b[e];
    g[e] = gv >= 0.f ? gv : 0.2f * gv;
    float nv = nbuf[b][e] + fc0_b[e];
    float s = fmaxf(nv, 0.f) + log1pf(expf(-fabsf(nv)));   // stable softplus
    sp[e] = s;
    nsum += s;
  }
  const float mu = nsum * (1.f / E_);
  float var = 0.f;
  #pragma unroll
  for (int e = 0; e < E_; ++e) { float d = sp[e] - mu; var += d * d; }
  const float inv_sd = rsqrtf(var * (1.f / (E_ - 1)));
  float score[E_];
  #pragma unroll
  for (int e = 0; e < E_; ++e) score[e] = g[e] + (sp[e] - mu) * inv_sd;

  bool sel[E_] = {false, false, false, false, false, false};
  for (int t = 0; t < KTOP; ++t) {
    float best = -INFINITY; int bi = 0;
    #pragma unroll
    for (int e = 0; e < E_; ++e)
      if (!sel[e] && score[e] > best) { best = score[e]; bi = e; }
    sel[bi] = true;
  }
  float gm = -INFINITY;
  #pragma unroll
  for (int e = 0; e < E_; ++e) if (sel[e]) gm = fmaxf(gm, g[e]);
  float den = 0.f, ex[E_];
  #pragma unroll
  for (int e = 0; e < E_; ++e) {
    ex[e] = sel[e] ? expf(g[e] - gm) : 0.f;
    den += ex[e];
  }
  const float inv = 1.f / den;
  #pragma unroll
  for (int e = 0; e < E_; ++e) cof[b * E_ + e] = ex[e] * inv;
}

// ---------------------------------------------------------------------------
// Kernel 3: fused 3-expert (dwconv3x3 -> relu -> dwconv3x3) * cof, summed.
// One (b,c) 32x64 tile per 256-thread block. x tile (36x68 + halo 2) staged
// into LDS (async global->LDS for interior tiles). Both conv phases use 2x4
// register micro-tiles: 24 LDS reads -> 8 results (3 reads/elem vs 9 naive).
// ---------------------------------------------------------------------------
__global__ __launch_bounds__(256) void mofe_conv_kernel(
    const float* __restrict__ x,
    const float* __restrict__ w1, const float* __restrict__ b1,
    const float* __restrict__ w2, const float* __restrict__ b2,
    const float* __restrict__ cof,
    float* __restrict__ out) {
  const int tx  = blockIdx.x;           // 0..2
  const int ty  = blockIdx.y;           // 0..5
  const int bc  = blockIdx.z;           // b*C + c
  const int b   = bc / C_;
  const int c   = bc % C_;
  const int gy0 = ty * TH;
  const int gx0 = tx * TW;
  const int tid = threadIdx.x;

  __shared__ float xs[XS_H * XS_W];     // padded stride 72
  __shared__ float hs[HS_H * HS_W];     // stride 67 (odd -> conflict-free)

  const float* __restrict__ xp = x + (size_t)bc * (H_ * W_);

  // ---- stage x tile (halo 2) into LDS ----
  const bool interior = (gy0 >= 2) && (gy0 + TH + 2 <= H_) &&
                        (gx0 >= 2) && (gx0 + TW + 2 <= W_);
  if (interior) {
#if defined(__gfx1250__)
    for (int idx = tid; idx < XS_H * XS_NW; idx += 256) {
      const int ly = idx / XS_NW, lx = idx % XS_NW;
      const float* gp = xp + (size_t)(gy0 - 2 + ly) * W_ + (gx0 - 2 + lx);
      // Flat LDS addresses keep the LDS byte offset in the low 32 bits.
      unsigned la = (unsigned)(unsigned long long)(const void*)&xs[ly * XS_W + lx];
      asm volatile("global_load_async_to_lds_b32 %0, %1, off"
                   :: "v"(la), "v"(gp) : "memory");
    }
    asm volatile("s_wait_asynccnt 0" ::: "memory");
#else
    for (int idx = tid; idx < XS_H * XS_NW; idx += 256) {
      const int ly = idx / XS_NW, lx = idx % XS_NW;
      xs[ly * XS_W + lx] = xp[(size_t)(gy0 - 2 + ly) * W_ + (gx0 - 2 + lx)];
    }
#endif
  } else {
    for (int idx = tid; idx < XS_H * XS_NW; idx += 256) {
      const int ly = idx / XS_NW, lx = idx % XS_NW;
      const int gy = gy0 - 2 + ly, gx = gx0 - 2 + lx;
      float v = 0.f;                                  // conv zero padding
      if (gy >= 0 && gy < H_ && gx >= 0 && gx < W_)
        v = xp[(size_t)gy * W_ + gx];
      xs[ly * XS_W + lx] = v;
    }
  }
  __syncthreads();

  float acc[8] = {0.f, 0.f, 0.f, 0.f, 0.f, 0.f, 0.f, 0.f};
  float cofs[E_];
  #pragma unroll
  for (int e = 0; e < E_; ++e) cofs[e] = cof[b * E_ + e];

  // phase-B micro-tile coordinates (fixed per thread)
  const int rgB = tid >> 4;             // 0..15  -> output rows 2*rgB, 2*rgB+1
  const int csB = tid & 15;             // 0..15  -> output cols 4*csB..+3
  const int oy0 = rgB * 2;
  const int ox0 = csB * 4;

  for (int e = 0; e < E_; ++e) {
    const float ce = cofs[e];
    if (ce == 0.0f) continue;           // uniform per block (b is uniform)

    const float* w1p = w1 + ((size_t)e * C_ + c) * 9;
    const float* w2p = w2 + ((size_t)e * C_ + c) * 9;
    float w1r[9], w2r[9];
    #pragma unroll
    for (int k = 0; k < 9; ++k) { w1r[k] = w1p[k]; w2r[k] = w2p[k]; }
    const float b1r = b1[e * C_ + c];
    const float b2r = b2[e * C_ + c];

    // ---- phase A: hs = relu(dwconv1(x)), 2x4 micro-tiles, 34x66 extent ----
    for (int s = tid; s < 17 * 17; s += 256) {
      const int rg = s / 17, cs = s % 17;
      const int ly0 = rg * 2, lx0 = cs * 4;
      float win[4][6];
      #pragma unroll
      for (int r = 0; r < 4; ++r)
        #pragma unroll
        for (int cc = 0; cc < 6; ++cc)
          win[r][cc] = xs[(ly0 + r) * XS_W + (lx0 + cc)];
      #pragma unroll
      for (int r = 0; r < 2; ++r) {
        #pragma unroll
        for (int cc = 0; cc < 4; ++cc) {
          const int hy = ly0 + r, hx = lx0 + cc;      // hs coords
          if (hx < HS_NW) {
            const int gy = gy0 - 1 + hy, gx = gx0 - 1 + hx;
            float v = 0.f;
            if (gy >= 0 && gy < H_ && gx >= 0 && gx < W_) {
              float sum = b1r;
              #pragma unroll
              for (int dy = 0; dy < 3; ++dy)
                #pragma unroll
                for (int dx = 0; dx < 3; ++dx)
                  sum = fmaf(w1r[dy * 3 + dx], win[r + dy][cc + dx], sum);
              v = fmaxf(sum, 0.f);
            }
            hs[hy * HS_W + hx] = v;
          }
        }
      }
    }
    __syncthreads();

    // ---- phase B: acc += cof * dwconv2(hs), 2x4 micro-tile per thread ----
    {
      float win[4][6];
      #pragma unroll
      for (int r = 0; r < 4; ++r)
        #pragma unroll
        for (int cc = 0; cc < 6; ++cc)
          win[r][cc] = hs[(oy0 + r) * HS_W + (ox0 + cc)];
      #pragma unroll
      for (int r = 0; r < 2; ++r) {
        #pragma unroll
        for (int cc = 0; cc < 4; ++cc) {
          float sum = b2r;
          #pragma unroll
          for (int dy = 0; dy < 3; ++dy)
            #pragma unroll
            for (int dx = 0; dx < 3; ++dx)
              sum = fmaf(w2r[dy * 3 + dx], win[r + dy][cc + dx], sum);
          acc[r * 4 + cc] = fmaf(ce, sum, acc[r * 4 + cc]);
        }
      }
    }
    __syncthreads();                    // hs reused by next expert
  }

  // ---- write 2x4 outputs as two aligned nontemporal 128-bit stores ----
  float* op = out + (size_t)bc * (H_ * W_) + (size_t)(gy0 + oy0) * W_ + (gx0 + ox0);
  v4f v0 = {acc[0], acc[1], acc[2], acc[3]};
  v4f v1 = {acc[4], acc[5], acc[6], acc[7]};
  __builtin_nontemporal_store(v0, (v4f*)op);
  __builtin_nontemporal_store(v1, (v4f*)(op + W_));
}

// ---------------------------------------------------------------------------
extern "C" void kernel_launch(void* const* d_in, const int* in_sizes, int n_in,
                              void* d_out, int out_size, void* d_ws, size_t ws_size,
                              hipStream_t stream) {
  const float* x     = (const float*)d_in[0];
  const float* fc0_w = (const float*)d_in[1];
  const float* fc0_b = (const float*)d_in[2];
  const float* fc1_w = (const float*)d_in[3];
  const float* fc1_b = (const float*)d_in[4];
  const float* w1    = (const float*)d_in[5];
  const float* b1    = (const float*)d_in[6];
  const float* w2    = (const float*)d_in[7];
  const float* b2    = (const float*)d_in[8];
  float* out = (float*)d_out;

  float* pooled = (float*)d_ws;           // B*C = 2048 floats
  float* cof    = pooled + B_ * C_;       // B*E = 192 floats

  mofe_pool_kernel<<<B_ * C_, 256, 0, stream>>>(x, pooled);
  mofe_gate_kernel<<<1, 32, 0, stream>>>(pooled, fc0_w, fc0_b, fc1_w, fc1_b, cof);
  dim3 grid(W_ / TW, H_ / TH, B_ * C_);   // 3 x 6 x 2048
  mofe_conv_kernel<<<grid, 256, 0, stream>>>(x, w1, b1, w2, b2, cof, out);
}